// NeuralMemory_24378234372436
// MI455X (gfx1250) — compile-verified
//
#include <hip/hip_runtime.h>
#include <math.h>

#define LEN 128
#define DIMN 256

typedef float v2f __attribute__((ext_vector_type(2)));
typedef float v8f __attribute__((ext_vector_type(8)));

// ---- f32 WMMA 16x16x4 wrapper (V_WMMA_F32_16X16X4_F32) ----
__device__ __forceinline__ v8f wmma4(v2f a, v2f b, v8f c) {
  return __builtin_amdgcn_wmma_f32_16x16x4_f32(false, a, false, b, (short)0, c,
                                               false, false);
}

__device__ __forceinline__ float sp(float x) { return x > 20.f ? x : log1pf(expf(x)); }
__device__ __forceinline__ float sigm(float x) { return 1.f / (1.f + expf(-x)); }
__device__ __forceinline__ float siluf(float x) { return x * sigm(x); }
__device__ __forceinline__ float silubwd(float z) {
  float s = sigm(z); float si = z * s; return si + s * (1.f - si);
}

// A-operand frag (16x4, rows r0.., cols k0..k0+3) — also B-operand for A@B^T
// lanes 0-15 hold K=k0,k0+1 ; lanes 16-31 hold K=k0+2,k0+3 (ISA 16x4 f32 layout)
__device__ __forceinline__ v2f fragT(const float* P, int r0, int k0, int ld, int lane) {
  int half = lane >> 4; int r = r0 + (lane & 15);
  const float* p = P + r * ld + k0 + 2 * half;
  v2f f; f.x = p[0]; f.y = p[1]; return f;
}
// B-operand frag for A@B (B row-major KxN)
__device__ __forceinline__ v2f fragN(const float* P, int k0, int c0, int ld, int lane) {
  int half = lane >> 4; int cc = c0 + (lane & 15); int k = k0 + 2 * half;
  v2f f; f.x = P[k * ld + cc]; f.y = P[(k + 1) * ld + cc]; return f;
}
// C = A(MxK) @ B^T  (B stored NxK row-major)
__device__ __forceinline__ v8f accT(const float* A, const float* B, int tm, int tn,
                                    int K, int lda, int ldb, int lane) {
  v8f acc = {};
  for (int k0 = 0; k0 < K; k0 += 4)
    acc = wmma4(fragT(A, tm * 16, k0, lda, lane), fragT(B, tn * 16, k0, ldb, lane), acc);
  return acc;
}
// C = A(MxK) @ B(KxN)
__device__ __forceinline__ v8f accN(const float* A, const float* B, int tm, int tn,
                                    int K, int lda, int ldb, int lane) {
  v8f acc = {};
  for (int k0 = 0; k0 < K; k0 += 4)
    acc = wmma4(fragT(A, tm * 16, k0, lda, lane), fragN(B, k0, tn * 16, ldb, lane), acc);
  return acc;
}
__device__ __forceinline__ void storeT(float* C, int r0, int c0, int ldc, int lane, v8f acc) {
  int half = lane >> 4; int cc = c0 + (lane & 15);
#pragma unroll
  for (int j = 0; j < 8; ++j) C[(r0 + j + 8 * half) * ldc + cc] = acc[j];
}

// ---- exact replication of the reference's nonstandard Blelloch scan as a
// 128x128 linear operator: column t = scan(a, e_t). Up-sweep of a basis vector
// stays single-nonzero (position p_d = t>>d, coefficient kappa_d), down-sweep:
// y[0]=b[0]; y[2i+1]=s[i]; y[2i]=b[2i]+a_d[2i]*s[i-1].
__device__ void build_M(const float* gates, float* alev, float* Mout, float scale, int t) {
  alev[t] = gates[t];
  __syncthreads();
  int off = 0, len = 128;
  for (int d = 0; d < 7; ++d) {
    int noff = off + len, half = len >> 1;
    if (t < half) alev[noff + t] = alev[off + 2 * t + 1] * alev[off + 2 * t];
    __syncthreads();
    off = noff; len = half;
  }
  float kappa[8]; kappa[0] = 1.f;
  {
    int p = t, o = 0, l = 128; float kv = 1.f;
    for (int d = 0; d < 7; ++d) {
      if ((p & 1) == 0) kv *= alev[o + p];
      p >>= 1; o += l; l >>= 1;
      kappa[d + 1] = kv;
    }
  }
  float ya[128], yb[128];
  float* cur = ya; float* nxt = yb;
  cur[0] = kappa[7];
  const int offs[7] = {0, 128, 192, 224, 240, 248, 252};
  for (int d = 6; d >= 0; --d) {
    int lenc = 64 >> d;           // len of cur (next level)
    int pd = t >> d; float kd = kappa[d];
    nxt[0] = (pd == 0) ? kd : 0.f;
    for (int i = 1; i < lenc; ++i)
      nxt[2 * i] = ((pd == 2 * i) ? kd : 0.f) + alev[offs[d] + 2 * i] * cur[i - 1];
    for (int i = 0; i < lenc; ++i) nxt[2 * i + 1] = cur[i];
    float* tp = cur; cur = nxt; nxt = tp;
  }
  for (int l = 0; l < 128; ++l) Mout[l * 128 + t] = cur[l] * scale;
  __syncthreads();
}

// ---- kernel 0: per-step scalars + scan operators ----
__global__ void k_prep(const float* x, const float* w_lr, const float* b_lr,
                       const float* w_mom, const float* b_mom,
                       const float* w_wd, const float* b_wd,
                       float* lr, float* mcum, float* wcum, float* cvec,
                       float* Mw, float* Mmlr) {
  __shared__ float g_fm[LEN], g_wd[LEN], s_lm[LEN], s_lw[LEN], alev[256];
  int t = threadIdx.x;  // 128 threads
  float dl = 0.f, dm = 0.f, dw = 0.f;
  const float* xr = x + t * DIMN;
  for (int d = 0; d < DIMN; ++d) {
    float xv = xr[d];
    dl += xv * w_lr[d]; dm += xv * w_mom[d]; dw += xv * w_wd[d];
  }
  float fm = dm + b_mom[0];
  float wdv = dw + b_wd[0];
  float LRC = logf(expm1f(0.01f));
  float lrv = sp(dl + b_lr[0] + LRC);
  lr[t] = lrv;
  g_fm[t] = fm; g_wd[t] = wdv;
  s_lm[t] = -sp(-fm); s_lw[t] = -sp(wdv);
  __syncthreads();
  if (t == 0) {
    float am = 0.f, aw = 0.f;
    for (int l = 0; l < LEN; ++l) {
      am += s_lm[l]; aw += s_lw[l];
      mcum[l] = expf(am); wcum[l] = expf(aw);
    }
  }
  __syncthreads();
  build_M(g_fm, alev, Mmlr, lrv, t);  // Mm_lr[l][k] = M_m[l][k]*lr[k]
  build_M(g_wd, alev, Mw, 1.f, t);
  __threadfence();
  __syncthreads();
  float acc = 0.f;
  for (int j = 0; j < LEN; ++j) acc += Mw[t * 128 + j] * mcum[j];
  cvec[t] = acc;
}

// ---- q,k,v projections ----
__global__ void k_qkv(const float* x, const float* Wq, const float* bq,
                      const float* Wk, const float* bk, const float* Wv, const float* bv,
                      float* q, float* key, float* v) {
  int wave = (blockIdx.x * blockDim.x + threadIdx.x) >> 5;
  int lane = threadIdx.x & 31;
  int which = wave >> 7, tile = wave & 127;
  int tm = tile >> 4, tn = tile & 15;
  const float* W = which == 0 ? Wq : (which == 1 ? Wk : Wv);
  const float* bias = which == 0 ? bq : (which == 1 ? bk : bv);
  float* out = which == 0 ? q : (which == 1 ? key : v);
  v8f acc = accT(x, W, tm, tn, DIMN, DIMN, DIMN, lane);
  int half = lane >> 4, cc = tn * 16 + (lane & 15);
  float bb = bias[cc];
#pragma unroll
  for (int j = 0; j < 8; ++j) out[(tm * 16 + j + 8 * half) * DIMN + cc] = acc[j] + bb;
}

// ---- Atilde = Mw @ Mm_lr ----
__global__ void k_atilde(const float* Mw, const float* Mmlr, float* At) {
  int wave = (blockIdx.x * blockDim.x + threadIdx.x) >> 5, lane = threadIdx.x & 31;
  int tm = wave >> 3, tn = wave & 7;
  v8f acc = accN(Mw, Mmlr, tm, tn, 128, 128, 128, lane);
  storeT(At, tm * 16, tn * 16, 128, lane, acc);
}

// ---- Z1 = key@W1^T + b1 ; X2 = silu(Z1) ----
__global__ void k_z1(const float* key, const float* W1, const float* b1,
                     float* Z1, float* X2) {
  int wave = (blockIdx.x * blockDim.x + threadIdx.x) >> 5, lane = threadIdx.x & 31;
  int tm = wave >> 4, tn = wave & 15;
  v8f acc = accT(key, W1, tm, tn, DIMN, DIMN, DIMN, lane);
  int half = lane >> 4, cc = tn * 16 + (lane & 15);
  float bb = b1[cc];
#pragma unroll
  for (int j = 0; j < 8; ++j) {
    int idx = (tm * 16 + j + 8 * half) * DIMN + cc;
    float z = acc[j] + bb;
    Z1[idx] = z; X2[idx] = siluf(z);
  }
}

// ---- gZ2 = X2@W2^T + b2 - v ----
__global__ void k_z2(const float* X2, const float* W2, const float* b2,
                     const float* v, float* gZ2) {
  int wave = (blockIdx.x * blockDim.x + threadIdx.x) >> 5, lane = threadIdx.x & 31;
  int tm = wave >> 4, tn = wave & 15;
  v8f acc = accT(X2, W2, tm, tn, DIMN, DIMN, DIMN, lane);
  int half = lane >> 4, cc = tn * 16 + (lane & 15);
  float bb = b2[cc];
#pragma unroll
  for (int j = 0; j < 8; ++j) {
    int idx = (tm * 16 + j + 8 * half) * DIMN + cc;
    gZ2[idx] = acc[j] + bb - v[idx];
  }
}

// ---- gZ1 = (gZ2@W2) * silu'(Z1) ----
__global__ void k_gz1(const float* gZ2, const float* W2, const float* Z1, float* gZ1) {
  int wave = (blockIdx.x * blockDim.x + threadIdx.x) >> 5, lane = threadIdx.x & 31;
  int tm = wave >> 4, tn = wave & 15;
  v8f acc = accN(gZ2, W2, tm, tn, DIMN, DIMN, DIMN, lane);
  int half = lane >> 4, cc = tn * 16 + (lane & 15);
#pragma unroll
  for (int j = 0; j < 8; ++j) {
    int idx = (tm * 16 + j + 8 * half) * DIMN + cc;
    gZ1[idx] = acc[j] * silubwd(Z1[idx]);
  }
}

// ---- P = At ⊙ (A@B^T + 1)   (128x128, K=256) ----
__global__ void k_pmat(const float* Aq, const float* Bk, const float* At, float* P) {
  int wave = (blockIdx.x * blockDim.x + threadIdx.x) >> 5, lane = threadIdx.x & 31;
  int tm = wave >> 3, tn = wave & 7;
  v8f acc = accT(Aq, Bk, tm, tn, DIMN, DIMN, DIMN, lane);
  int half = lane >> 4, cc = tn * 16 + (lane & 15);
#pragma unroll
  for (int j = 0; j < 8; ++j) {
    int r = tm * 16 + j + 8 * half;
    P[r * 128 + cc] = At[r * 128 + cc] * (acc[j] + 1.f);
  }
}

// ---- Zq = P@G + wcum⊙(A2@W0^T + b0) - c⊙(A2@mW0^T + mb0) [optional silu] ----
__global__ void k_zq(const float* P, const float* G, const float* A2,
                     const float* W0, const float* mW0, const float* b0,
                     const float* mb0, const float* wcum, const float* cvec,
                     float* out, int do_silu) {
  int wave = (blockIdx.x * blockDim.x + threadIdx.x) >> 5, lane = threadIdx.x & 31;
  int tm = wave >> 4, tn = wave & 15;
  v8f a1 = accN(P, G, tm, tn, 128, 128, DIMN, lane);
  v8f a2 = accT(A2, W0, tm, tn, DIMN, DIMN, DIMN, lane);
  v8f a3 = accT(A2, mW0, tm, tn, DIMN, DIMN, DIMN, lane);
  int half = lane >> 4, cc = tn * 16 + (lane & 15);
  float bb = b0[cc], mbb = mb0[cc];
#pragma unroll
  for (int j = 0; j < 8; ++j) {
    int r = tm * 16 + j + 8 * half;
    float z = a1[j] + wcum[r] * (a2[j] + bb) - cvec[r] * (a3[j] + mbb);
    out[r * DIMN + cc] = do_silu ? siluf(z) : z;
  }
}

// ---- final weight states: outP = Σk af[k] G[k,m] B[k,n] + wcf*W0 - cf*mW0
//                           outM = Σk bf[k] G[k,m] B[k,n] - mcf*mW0 ----
__global__ void k_final(const float* G, const float* Bm, const float* At,
                        const float* Mmlr, const float* W0, const float* mW0,
                        const float* wcum, const float* mcum, const float* cvec,
                        float* outP, float* outM) {
  int wave = (blockIdx.x * blockDim.x + threadIdx.x) >> 5, lane = threadIdx.x & 31;
  int tm = wave >> 4, tn = wave & 15;  // 16x16 tiles over 256x256
  const float* af = At + 127 * 128;
  const float* bf = Mmlr + 127 * 128;
  int half = lane >> 4, m = tm * 16 + (lane & 15);
  v8f accA = {}, accB = {};
  for (int k0 = 0; k0 < 128; k0 += 4) {
    int k = k0 + 2 * half;
    float g0 = G[k * DIMN + m], g1 = G[(k + 1) * DIMN + m];
    v2f fa; fa.x = af[k] * g0; fa.y = af[k + 1] * g1;
    v2f fb; fb.x = bf[k] * g0; fb.y = bf[k + 1] * g1;
    v2f bbf = fragN(Bm, k0, tn * 16, DIMN, lane);
    accA = wmma4(fa, bbf, accA);
    accB = wmma4(fb, bbf, accB);
  }
  float wcf = wcum[127], mcf = mcum[127], cf = cvec[127];
  int cc = tn * 16 + (lane & 15);
#pragma unroll
  for (int j = 0; j < 8; ++j) {
    int idx = (tm * 16 + j + 8 * half) * DIMN + cc;
    outP[idx] = accA[j] + wcf * W0[idx] - cf * mW0[idx];
    outM[idx] = accB[j] - mcf * mW0[idx];
  }
}

// ---- final bias states ----
__global__ void k_bfinal(const float* gZ1, const float* gZ2, const float* At,
                         const float* Mmlr, const float* b1, const float* mb1,
                         const float* b2, const float* mb2, const float* wcum,
                         const float* mcum, const float* cvec,
                         float* b1p, float* mb1o, float* b2p, float* mb2o) {
  int h = threadIdx.x;  // 256
  const float* af = At + 127 * 128;
  const float* bf = Mmlr + 127 * 128;
  float s1 = 0.f, s2 = 0.f, s3 = 0.f, s4 = 0.f;
  for (int k = 0; k < 128; ++k) {
    float g1 = gZ1[k * DIMN + h], g2 = gZ2[k * DIMN + h];
    s1 += af[k] * g1; s2 += bf[k] * g1;
    s3 += af[k] * g2; s4 += bf[k] * g2;
  }
  float wcf = wcum[127], mcf = mcum[127], cf = cvec[127];
  b1p[h] = s1 - cf * mb1[h] + wcf * b1[h];
  mb1o[h] = s2 - mcf * mb1[h];
  b2p[h] = s3 - cf * mb2[h] + wcf * b2[h];
  mb2o[h] = s4 - mcf * mb2[h];
}

extern "C" void kernel_launch(void* const* d_in, const int* in_sizes, int n_in,
                              void* d_out, int out_size, void* d_ws, size_t ws_size,
                              hipStream_t stream) {
  const float* x   = (const float*)d_in[0];
  const float* W1  = (const float*)d_in[1];
  const float* b1  = (const float*)d_in[2];
  const float* W2  = (const float*)d_in[3];
  const float* b2  = (const float*)d_in[4];
  const float* mW1 = (const float*)d_in[5];
  const float* mb1 = (const float*)d_in[6];
  const float* mW2 = (const float*)d_in[7];
  const float* mb2 = (const float*)d_in[8];
  const float* Wq  = (const float*)d_in[9];
  const float* bq  = (const float*)d_in[10];
  const float* Wk  = (const float*)d_in[11];
  const float* bk  = (const float*)d_in[12];
  const float* Wv  = (const float*)d_in[13];
  const float* bv  = (const float*)d_in[14];
  const float* w_lr  = (const float*)d_in[15];
  const float* b_lr  = (const float*)d_in[16];
  const float* w_mom = (const float*)d_in[17];
  const float* b_mom = (const float*)d_in[18];
  const float* w_wd  = (const float*)d_in[19];
  const float* b_wd  = (const float*)d_in[20];
  float* out = (float*)d_out;

  float* W = (float*)d_ws;
  float* lr    = W + 0;        // 128
  float* mcum  = W + 128;      // 128
  float* wcum  = W + 256;      // 128
  float* cvec  = W + 384;      // 128
  float* Mw    = W + 512;      // 16384
  float* Mmlr  = W + 16896;    // 16384
  float* At    = W + 33280;    // 16384
  float* q     = W + 49664;    // 32768
  float* key   = W + 82432;    // 32768
  float* v     = W + 115200;   // 32768
  float* Z1    = W + 147968;   // 32768
  float* X2    = W + 180736;   // 32768
  float* gZ2   = W + 213504;   // 32768
  float* gZ1   = W + 246272;   // 32768
  float* Xq    = W + 279040;   // 32768
  float* P1    = W + 311808;   // 16384
  float* P2    = W + 328192;   // 16384

  // output tuple offsets (floats)
  float* o_Zq2  = out;            // 32768
  float* o_W1p  = out + 32768;    // 65536
  float* o_b1p  = out + 98304;    // 256
  float* o_W2p  = out + 98560;    // 65536
  float* o_b2p  = out + 164096;   // 256
  float* o_mW1  = out + 164352;   // 65536
  float* o_mb1  = out + 229888;   // 256
  float* o_mW2  = out + 230144;   // 65536
  float* o_mb2  = out + 295680;   // 256

  k_prep<<<1, 128, 0, stream>>>(x, w_lr, b_lr, w_mom, b_mom, w_wd, b_wd,
                                lr, mcum, wcum, cvec, Mw, Mmlr);
  k_qkv<<<48, 256, 0, stream>>>(x, Wq, bq, Wk, bk, Wv, bv, q, key, v);
  k_atilde<<<8, 256, 0, stream>>>(Mw, Mmlr, At);
  k_z1<<<16, 256, 0, stream>>>(key, W1, b1, Z1, X2);
  k_z2<<<16, 256, 0, stream>>>(X2, W2, b2, v, gZ2);
  k_gz1<<<16, 256, 0, stream>>>(gZ2, W2, Z1, gZ1);
  k_pmat<<<8, 256, 0, stream>>>(q, key, At, P1);
  k_zq<<<16, 256, 0, stream>>>(P1, gZ1, q, W1, mW1, b1, mb1, wcum, cvec, Xq, 1);
  k_pmat<<<8, 256, 0, stream>>>(Xq, X2, At, P2);
  k_zq<<<16, 256, 0, stream>>>(P2, gZ2, Xq, W2, mW2, b2, mb2, wcum, cvec, o_Zq2, 0);
  k_final<<<32, 256, 0, stream>>>(gZ1, key, At, Mmlr, W1, mW1, wcum, mcum, cvec,
                                  o_W1p, o_mW1);
  k_final<<<32, 256, 0, stream>>>(gZ2, X2, At, Mmlr, W2, mW2, wcum, mcum, cvec,
                                  o_W2p, o_mW2);
  k_bfinal<<<1, 256, 0, stream>>>(gZ1, gZ2, At, Mmlr, b1, mb1, b2, mb2,
                                  wcum, mcum, cvec, o_b1p, o_mb1, o_b2p, o_mb2);
  (void)in_sizes; (void)n_in; (void)out_size; (void)ws_size;
}